// GCNLayer_91250875171357
// MI455X (gfx1250) — compile-verified
//
#include <hip/hip_runtime.h>
#include <stdint.h>

// GCN aggregation: out[d] = in_dg[d] * sum_{(s->d) in E} node_f[s] * out_d[s]
// in_dg folded per-edge (linearity of the sum).
//
// MI455X plan:
//  * node_f (12.8MB), out (12.8MB), degree vecs (0.8MB) are L2-resident (192MB L2);
//    gather + f32 atomic scatter resolve at the L2 atomic units.
//  * Edge lists (12.8MB) stream ONCE from HBM via CDNA5 async DMA:
//    GLOBAL_LOAD_ASYNC_TO_LDS_B64, double-buffered per wave, synced with
//    s_wait_asynccnt; inner loop reads indices as ds_load_b128 broadcasts.
//  * Indices are wave-uniform -> v_readfirstlane scalarizes them: degree loads
//    become SMEM s_load_b32 (scalar path), gather/atomic use saddr+lane form,
//    index math moves to SALU.
//  * wave32: one wave per edge per inner step => 128B coalesced row gather +
//    128B coalesced global_atomic_add_f32 burst; 32 atomics in flight per wave.

#define D_FEAT 32
#define CHUNK  32   // edges per staged chunk: 128B src + 128B dst = 256B per async op
#define WAVES_PER_BLOCK 8

__device__ __forceinline__ void async_stage_b64(uint32_t lds_off, const int* gaddr) {
    // VDST = per-lane LDS byte offset (workgroup-relative), VADDR = 64-bit global addr.
    asm volatile("global_load_async_to_lds_b64 %0, %1, off"
                 :: "v"(lds_off), "v"(gaddr)
                 : "memory");
}
__device__ __forceinline__ void wait_async_le1() {
    asm volatile("s_wait_asynccnt 0x1" ::: "memory");
}
__device__ __forceinline__ void wait_async_le0() {
    asm volatile("s_wait_asynccnt 0x0" ::: "memory");
}

__global__ __launch_bounds__(256)
void GCN_zero_kernel(float4* __restrict__ out, int n4) {
    int i = blockIdx.x * blockDim.x + threadIdx.x;
    if (i < n4) out[i] = make_float4(0.f, 0.f, 0.f, 0.f);
}

__global__ __launch_bounds__(256)
void GCN_scatter_async(const float* __restrict__ node_f,
                       const float* __restrict__ out_d,
                       const float* __restrict__ in_dg,
                       const int*   __restrict__ src,
                       const int*   __restrict__ dst,
                       float*       __restrict__ out,
                       int n_edges) {
    // [wave-in-block][buffer][0..CHUNK-1 = src | CHUNK..2*CHUNK-1 = dst]
    __shared__ int sbuf[WAVES_PER_BLOCK][2][2 * CHUNK];

    const int tid    = blockIdx.x * blockDim.x + threadIdx.x;
    const int wave   = tid >> 5;
    const int lane   = threadIdx.x & 31;
    const int wid    = threadIdx.x >> 5;
    const int nwaves = (gridDim.x * blockDim.x) >> 5;
    const int n_full = n_edges / CHUNK;

    // Staging lane roles: lanes 0-15 copy src (8B each), lanes 16-31 copy dst.
    const int  li     = lane & 15;
    const bool is_dst = (lane >= 16);

    auto stage = [&](int buf, int base_edge) {
        const int* g = (is_dst ? dst : src) + base_edge + li * 2;      // 2 ints = 8B
        // Low 32 bits of a generic __shared__ pointer == workgroup-relative LDS offset.
        uint32_t loff = (uint32_t)(uintptr_t)(&sbuf[wid][buf][0])
                      + (is_dst ? (uint32_t)(CHUNK * 4) : 0u)
                      + (uint32_t)(li * 8);
        async_stage_b64(loff, g);
    };

    int c   = wave;
    int buf = 0;
    if (c < n_full) stage(0, c * CHUNK);

    for (; c < n_full; c += nwaves) {
        const int cn = c + nwaves;
        if (cn < n_full) {          // keep the pipe full: issue next, wait for current
            stage(buf ^ 1, cn * CHUNK);
            wait_async_le1();       // in-order completion => chunk c is in LDS
        } else {
            wait_async_le0();
        }

        const int* s_idx = &sbuf[wid][buf][0];
        const int* d_idx = &sbuf[wid][buf][CHUNK];
        #pragma unroll 8
        for (int i = 0; i < CHUNK; ++i) {
            // All lanes read identical LDS words -> assert wave-uniformity so the
            // compiler uses SMEM for the degree vectors and saddr-form VMEM for
            // the gather/scatter (index math on the scalar pipe).
            const int s = __builtin_amdgcn_readfirstlane(s_idx[i]);
            const int d = __builtin_amdgcn_readfirstlane(d_idx[i]);
            const float w = out_d[s] * in_dg[d];    // s_load_b32 (L2-resident)
            const float v = node_f[(size_t)s * D_FEAT + lane] * w;
            unsafeAtomicAdd(out + (size_t)d * D_FEAT + lane, v);  // global_atomic_add_f32
        }
        buf ^= 1;
    }

    // Tail edges (n_edges % CHUNK): direct path, one edge per wave.
    const int e = n_full * CHUNK + wave;
    if (e < n_edges) {
        const int s = src[e];
        const int d = dst[e];
        const float w = out_d[s] * in_dg[d];
        const float v = node_f[(size_t)s * D_FEAT + lane] * w;
        unsafeAtomicAdd(out + (size_t)d * D_FEAT + lane, v);
    }
}

extern "C" void kernel_launch(void* const* d_in, const int* in_sizes, int n_in,
                              void* d_out, int out_size, void* d_ws, size_t ws_size,
                              hipStream_t stream) {
    const float* node_f = (const float*)d_in[0];   // [N, 32] f32
    const float* out_d  = (const float*)d_in[1];   // [N, 1]  f32
    const float* in_dg  = (const float*)d_in[2];   // [N, 1]  f32
    const int*   src    = (const int*)d_in[3];     // [E]     i32
    const int*   dst    = (const int*)d_in[4];     // [E]     i32
    float*       out    = (float*)d_out;           // [N, 32] f32

    const int n_edges = in_sizes[3];

    // 1) Zero the accumulator every call (harness poisons d_out once, never re-poisons).
    const int n4 = out_size / 4;                   // out_size = N*32, divisible by 4
    GCN_zero_kernel<<<(n4 + 255) / 256, 256, 0, stream>>>((float4*)out, n4);

    // 2) Async-pipelined edge scatter: 1024 blocks x 8 waves = 8192 waves,
    //    ~6 chunks/wave over E/32 = 50k chunks -> double buffer stays busy.
    GCN_scatter_async<<<1024, 256, 0, stream>>>(node_f, out_d, in_dg, src, dst, out,
                                                n_edges);
}